// GraphAttentionLayer_85933705658412
// MI455X (gfx1250) — compile-verified
//
#include <hip/hip_runtime.h>
#include <hip/hip_bf16.h>

#define F 64  // channels == F_IN == 64

typedef __attribute__((ext_vector_type(2))) float v2f;
typedef __attribute__((ext_vector_type(8))) float v8f;

// ---- order-preserving float <-> uint mapping for atomic max on floats ----
__device__ __forceinline__ unsigned f2o(float f) {
    unsigned u = __float_as_uint(f);
    return (u & 0x80000000u) ? ~u : (u | 0x80000000u);
}
__device__ __forceinline__ float o2f(unsigned u) {
    return (u & 0x80000000u) ? __uint_as_float(u & 0x7FFFFFFFu) : __uint_as_float(~u);
}

// ------------------------------------------------------------------
// Kernel 1: h = x @ W via V_WMMA_F32_16X16X4_F32 (full fp32 WMMA).
// One wave computes one 16x16 output tile; 16 WMMAs chained over K=64.
// A 16x4 layout : lanes 0-15 -> M=lane, K=k+{0,1}; lanes 16-31 -> K=k+{2,3}
// B 4x16 layout : lanes 0-15 -> N=lane, K=k+{0,1}; lanes 16-31 -> K=k+{2,3}
// D 16x16 layout: VGPR r -> M=r (lanes 0-15) / M=r+8 (lanes 16-31), N=lane&15
// ------------------------------------------------------------------
__global__ __launch_bounds__(256) void gat_gemm_wmma(
    const float* __restrict__ x, const float* __restrict__ W,
    float* __restrict__ h, int N) {
    int wave = blockIdx.x * (blockDim.x >> 5) + (threadIdx.x >> 5);
    int lane = threadIdx.x & 31;
    int row  = (wave >> 2) << 4;   // 16-row tile
    int col  = (wave & 3) << 4;    // 16-col tile
    if (row >= N) return;          // wave-uniform exit (EXEC stays all-1s)

    int mn    = lane & 15;         // M for A loads, N for B loads / D stores
    int khalf = (lane >> 4) << 1;  // 0 for lanes 0-15, 2 for lanes 16-31

    int arow = row + mn; if (arow > N - 1) arow = N - 1;  // clamp (defensive)
    const float* xrow = x + (size_t)arow * F;

    v8f acc = {};
#pragma unroll
    for (int k = 0; k < F; k += 4) {
        float2 av = *(const float2*)(xrow + k + khalf);
        v2f a; a.x = av.x; a.y = av.y;
        v2f b;
        b.x = W[(size_t)(k + khalf)     * F + col + mn];
        b.y = W[(size_t)(k + khalf + 1) * F + col + mn];
        acc = __builtin_amdgcn_wmma_f32_16x16x4_f32(
            /*neg_a=*/false, a, /*neg_b=*/false, b,
            /*c_mod=*/(short)0, acc, /*reuse_a=*/false, /*reuse_b=*/false);
    }

    int mo = (lane >> 4) << 3;  // 0 or 8
#pragma unroll
    for (int r = 0; r < 8; ++r) {
        int m = row + mo + r;
        if (m < N) h[(size_t)m * F + col + mn] = acc[r];
    }
}

// Kernel 2: per-node attention scores s_src = h.a_src, s_dst = h.a_dst
__global__ void gat_scores(const float* __restrict__ h,
                           const float* __restrict__ a_src,
                           const float* __restrict__ a_dst,
                           float* __restrict__ s_src, float* __restrict__ s_dst,
                           int N) {
    int i = blockIdx.x * blockDim.x + threadIdx.x;
    if (i >= N) return;
    const float4* hv = (const float4*)(h + (size_t)i * F);
    const float4* as = (const float4*)a_src;
    const float4* ad = (const float4*)a_dst;
    float ss = 0.f, sd = 0.f;
#pragma unroll
    for (int c = 0; c < F / 4; ++c) {
        float4 hh = hv[c], s4 = as[c], d4 = ad[c];
        ss += hh.x * s4.x + hh.y * s4.y + hh.z * s4.z + hh.w * s4.w;
        sd += hh.x * d4.x + hh.y * d4.y + hh.z * d4.z + hh.w * d4.w;
    }
    s_src[i] = ss; s_dst[i] = sd;
}

// Kernel 3: init segment-max (ordered -inf), denom, accumulator
__global__ void gat_init(unsigned* __restrict__ m_ord, float* __restrict__ denom,
                         float* __restrict__ acc, int N, long total) {
    long i = (long)blockIdx.x * blockDim.x + threadIdx.x;
    if (i < N) { m_ord[i] = 0x007FFFFFu; /* f2o(-inf) */ denom[i] = 0.f; }
    if (i < total) acc[i] = 0.f;
}

// Kernel 4: edge logits + segment max
__global__ void gat_edge_max(const int* __restrict__ esrc, const int* __restrict__ edst,
                             const float* __restrict__ s_src, const float* __restrict__ s_dst,
                             float* __restrict__ e_buf, unsigned* __restrict__ m_ord, int E) {
    int e = blockIdx.x * blockDim.x + threadIdx.x;
    if (e >= E) return;
    int s = esrc[e], d = edst[e];
    float v = s_src[s] + s_dst[d];
    v = v > 0.f ? v : 0.2f * v;             // LeakyReLU(0.2)
    e_buf[e] = v;
    atomicMax(&m_ord[d], f2o(v));
}

// Kernel 5: ex = exp(e - m[dst]); denom[dst] += ex
__global__ void gat_edge_exp(const int* __restrict__ edst,
                             const float* __restrict__ e_buf,
                             const unsigned* __restrict__ m_ord,
                             float* __restrict__ ex_buf, float* __restrict__ denom, int E) {
    int e = blockIdx.x * blockDim.x + threadIdx.x;
    if (e >= E) return;
    int d = edst[e];
    float ex = expf(e_buf[e] - o2f(m_ord[d]));
    ex_buf[e] = ex;
    atomicAdd(&denom[d], ex);
}

// Kernel 6: acc[dst] += alpha * h[src]; 16 threads/edge, 4 channels each
__global__ void gat_scatter(const int* __restrict__ esrc, const int* __restrict__ edst,
                            const float* __restrict__ h, const float* __restrict__ ex_buf,
                            const float* __restrict__ denom, float* __restrict__ acc, int E) {
    long t = (long)blockIdx.x * blockDim.x + threadIdx.x;
    int e = (int)(t >> 4);
    int q = (int)(t & 15);
    if (e >= E) return;
    int s = esrc[e], d = edst[e];
    float dn = denom[d]; dn = dn > 1e-9f ? dn : 1e-9f;
    float alpha = ex_buf[e] / dn;
    float4 hv = ((const float4*)(h + (size_t)s * F))[q];
    float* ap = acc + (size_t)d * F + q * 4;
    atomicAdd(ap + 0, alpha * hv.x);
    atomicAdd(ap + 1, alpha * hv.y);
    atomicAdd(ap + 2, alpha * hv.z);
    atomicAdd(ap + 3, alpha * hv.w);
}

// Kernel 7: out = LeakyReLU(acc + b, 0.3)
__global__ void gat_final(const float* __restrict__ acc, const float* __restrict__ b,
                          float* __restrict__ out, long total) {
    long i = (long)blockIdx.x * blockDim.x + threadIdx.x;
    if (i >= total) return;
    float v = acc[i] + b[i & (F - 1)];
    out[i] = v > 0.f ? v : 0.3f * v;
}

extern "C" void kernel_launch(void* const* d_in, const int* in_sizes, int n_in,
                              void* d_out, int out_size, void* d_ws, size_t ws_size,
                              hipStream_t stream) {
    const float* x     = (const float*)d_in[0];
    const float* W     = (const float*)d_in[1];
    const float* a_src = (const float*)d_in[2];
    const float* a_dst = (const float*)d_in[3];
    const float* b     = (const float*)d_in[4];
    const int*   esrc  = (const int*)d_in[5];
    const int*   edst  = (const int*)d_in[6];

    const int N = in_sizes[0] / F;
    const int E = in_sizes[5];
    const long total = (long)N * F;

    // Carve workspace
    char* ws = (char*)d_ws;
    size_t off = 0;
    auto carve = [&](size_t bytes) -> void* {
        void* p = ws + off;
        off = (off + bytes + 255) & ~(size_t)255;
        return p;
    };
    float*    h      = (float*)carve((size_t)total * 4);
    float*    acc    = (float*)carve((size_t)total * 4);
    float*    s_src  = (float*)carve((size_t)N * 4);
    float*    s_dst  = (float*)carve((size_t)N * 4);
    unsigned* m_ord  = (unsigned*)carve((size_t)N * 4);
    float*    denom  = (float*)carve((size_t)N * 4);
    float*    e_buf  = (float*)carve((size_t)E * 4);
    float*    ex_buf = (float*)carve((size_t)E * 4);
    (void)ws_size; (void)n_in; (void)out_size;

    // 1) h = x @ W  (WMMA fp32)
    int rowTiles = (N + 15) / 16;
    int waves = rowTiles * 4;               // 4 col-tiles of 16
    int gemmBlocks = (waves + 7) / 8;       // 8 waves per 256-thread block
    gat_gemm_wmma<<<gemmBlocks, 256, 0, stream>>>(x, W, h, N);

    // 2) per-node scores
    gat_scores<<<(N + 255) / 256, 256, 0, stream>>>(h, a_src, a_dst, s_src, s_dst, N);

    // 3) init max/denom/acc
    gat_init<<<(int)((total + 255) / 256), 256, 0, stream>>>(m_ord, denom, acc, N, total);

    // 4) edge logits + segment max
    gat_edge_max<<<(E + 255) / 256, 256, 0, stream>>>(esrc, edst, s_src, s_dst, e_buf, m_ord, E);

    // 5) exp + segment sum
    gat_edge_exp<<<(E + 255) / 256, 256, 0, stream>>>(edst, e_buf, m_ord, ex_buf, denom, E);

    // 6) weighted scatter-aggregate
    long st = (long)E * 16;
    gat_scatter<<<(int)((st + 255) / 256), 256, 0, stream>>>(esrc, edst, h, ex_buf, denom, acc, E);

    // 7) bias + LeakyReLU(0.3)
    gat_final<<<(int)((total + 255) / 256), 256, 0, stream>>>(acc, b, (float*)d_out, total);
}